// Discriminator_39994735460557
// MI455X (gfx1250) — compile-verified
//
#include <hip/hip_runtime.h>
#include <hip/hip_bf16.h>

typedef float v2f __attribute__((ext_vector_type(2)));
typedef float v8f __attribute__((ext_vector_type(8)));

#define DIM   500
#define NSEC  11
#define NMQ   10
#define LDSTR 516          // 516 % 64 == 4 -> conflict-free LDS column reads
#define KG    125          // 500 / 4 K-groups
#define NPAD  512          // N padded to 32 tiles of 16
#define PACKN (KG * NPAD * 4)   // packed Omega: [kg][col][kk], 1 MB

__device__ __forceinline__ float relu_(float v) { return fmaxf(v, 0.0f); }

__device__ __forceinline__ float wred32(float v) {
#pragma unroll
    for (int o = 16; o > 0; o >>= 1) v += __shfl_xor(v, o, 32);
    return v;
}

// ---------------------------------------------------------------------------
// Kernel 0: pack Omega into WMMA B-fragment order.
// Bp[kg*2048 + col*4 + kk] = Omega[(4*kg+kk)*500 + col]  (0 if col >= 500).
// A lane's two K-values (ka, ka+1) become ONE aligned global_load_b64.
// ---------------------------------------------------------------------------
__global__ __launch_bounds__(256) void pack_omega_kernel(
    const float* __restrict__ Omega, float* __restrict__ Bp)
{
    const int idx = blockIdx.x * 256 + threadIdx.x;
    if (idx >= PACKN) return;
    const int kg   = idx >> 11;          // / 2048
    const int rem  = idx & 2047;
    const int col  = rem >> 2;
    const int kk   = rem & 3;
    const int krow = kg * 4 + kk;        // always < 500
    Bp[idx] = (col < DIM) ? Omega[(size_t)krow * DIM + col] : 0.0f;
}

// ---------------------------------------------------------------------------
// Kernel 1: per-row streaming stats (one wave per row, 8 rows/block).
// ---------------------------------------------------------------------------
__global__ __launch_bounds__(256) void row_stats_kernel(
    const float* __restrict__ x, const float* __restrict__ x_bw,
    const float* __restrict__ alpha, const float* __restrict__ beta,
    const float* __restrict__ sector, const float* __restrict__ mq,
    float* __restrict__ ws_tot, float* __restrict__ ws_l2,
    float* __restrict__ ws_sa, int B)
{
    const int wave = threadIdx.x >> 5;
    const int lane = threadIdx.x & 31;
    const int row  = blockIdx.x * 8 + wave;
    if (row >= B) return;

    const float* xr = x + (size_t)row * DIM;

    float s = 0.f, sa = 0.f, db = 0.f, da = 0.f, nnz = 0.f;
    float ks[NSEC]; float km[NMQ];
#pragma unroll
    for (int g = 0; g < NSEC; ++g) ks[g] = 0.f;
#pragma unroll
    for (int g = 0; g < NMQ; ++g) km[g] = 0.f;

    for (int k = lane; k < DIM; k += 32) {
        const float xv = xr[k];
        const float d  = xv - x_bw[k];
        s   += xv;
        sa  += fabsf(d);
        db  += d * beta[k];
        da  += d * alpha[k];
        nnz += tanhf(xv * 1000.0f);        // tanh(x / X_THRESH)
#pragma unroll
        for (int g = 0; g < NSEC; ++g) ks[g] += d * sector[g * DIM + k];
#pragma unroll
        for (int g = 0; g < NMQ; ++g)  km[g] += d * mq[g * DIM + k];
    }

    s = wred32(s); sa = wred32(sa); db = wred32(db); da = wred32(da); nnz = wred32(nnz);
#pragma unroll
    for (int g = 0; g < NSEC; ++g) ks[g] = wred32(ks[g]);
#pragma unroll
    for (int g = 0; g < NMQ; ++g)  km[g] = wred32(km[g]);

    if (lane == 0) {
        float tot = relu_(1.0f - s) + relu_(s - 1.0f);
        tot += relu_(sa - 0.05f);
#pragma unroll
        for (int g = 0; g < NSEC; ++g) tot += relu_(fabsf(ks[g]) - 0.1f);
#pragma unroll
        for (int g = 0; g < NMQ; ++g)  tot += relu_(fabsf(km[g]) - 0.1f);
        tot += relu_(fabsf(db) - 0.1f);
        tot += relu_(nnz - 70.0f) + relu_(50.0f - nnz);
        ws_tot[row] = tot;
        ws_l2[row]  = da;
        ws_sa[row]  = sa;
    }
}

// ---------------------------------------------------------------------------
// Kernel 2: deterministic global reduction of sum|d| -> l_scalar = 0.5*total.
// ---------------------------------------------------------------------------
__global__ __launch_bounds__(1024) void reduce_sa_kernel(
    const float* __restrict__ ws_sa, float* __restrict__ ws_scalar, int n)
{
    __shared__ float sh[1024];
    float s = 0.f;
    for (int i = threadIdx.x; i < n; i += 1024) s += ws_sa[i];
    sh[threadIdx.x] = s;
    __syncthreads();
#pragma unroll
    for (int o = 512; o > 0; o >>= 1) {
        if (threadIdx.x < o) sh[threadIdx.x] += sh[threadIdx.x + o];
        __syncthreads();
    }
    if (threadIdx.x == 0) ws_scalar[0] = 0.5f * sh[0];
}

// ---------------------------------------------------------------------------
// Kernel 3: dQd[b] = d_b^T * Omega * d_b via V_WMMA_F32_16X16X4_F32.
// Block = 256 threads (8 waves) handles 64 rows (four 16-row M-tiles) in
// dynamic LDS (132 KB of CDNA5's 320 KB WGP; 2 blocks co-resident).
// Wave grid: 2 (M-pairs) x 4 (N). Each wave owns TWO M-tiles and, per pass,
// TWO concurrent N-tiles: each K-step does 2 ds_load_b64 (A) + 2
// global_load_b64 (B, packed layout -> unconditional, aligned) feeding
// FOUR independent v_wmma chains, i.e. each B fragment is reused by two
// WMMAs. This doubles arithmetic intensity on the Omega stream vs. the
// previous blocking and halves total L2 re-stream traffic (block covers
// 64 rows). dQ tiles are contracted in-register against the LDS d-tile and
// combined across waves deterministically through LDS.
// ---------------------------------------------------------------------------
__global__ __launch_bounds__(256) void gemm_dqd_kernel(
    const float* __restrict__ x, const float* __restrict__ x_bw,
    const float* __restrict__ Bp, float* __restrict__ ws_dqd, int B)
{
    extern __shared__ float smem[];
    float* lds_d    = smem;                 // 64 * LDSTR floats
    float* lds_part = smem + 64 * LDSTR;    // 8 * 2 * 16 floats

    const int row0 = blockIdx.x * 64;

    // Cooperative load of d tile: d = x - x_bw, zero padded (cols 500..515).
    for (int idx = threadIdx.x; idx < 64 * LDSTR; idx += 256) {
        const int r = idx / LDSTR;
        const int c = idx % LDSTR;
        float v = 0.f;
        if (c < DIM && (row0 + r) < B) v = x[(size_t)(row0 + r) * DIM + c] - x_bw[c];
        lds_d[idx] = v;
    }
    __syncthreads();

    const int wave = threadIdx.x >> 5;
    const int lane = threadIdx.x & 31;
    const int mp   = wave >> 2;       // M-pair 0/1 -> M-tiles 2*mp, 2*mp+1
    const int wsub = wave & 3;        // N-column of the wave grid
    const int half = lane >> 4;       // 0: lanes 0-15, 1: lanes 16-31
    const int l15  = lane & 15;

    const float* arow0 = &lds_d[((2 * mp) * 16 + l15) * LDSTR];
    const float* arow1 = &lds_d[((2 * mp + 1) * 16 + l15) * LDSTR];

    float acc0[8], acc1[8];
#pragma unroll
    for (int r = 0; r < 8; ++r) { acc0[r] = 0.f; acc1[r] = 0.f; }

#pragma unroll 1
    for (int p = 0; p < 4; ++p) {     // 4 passes of 2 concurrent N-tiles
        const int col0 = (wsub + 8 * p) * 16 + l15;
        const int col1 = (wsub + 8 * p + 4) * 16 + l15;

        v8f c00 = {}, c01 = {}, c10 = {}, c11 = {};
        for (int k = 0; k < DIM; k += 4) {        // 500 = 4 * 125, exact
            const int kg = k >> 2;
            const v2f a0 = *(const v2f*)&arow0[k + (half << 1)];
            const v2f a1 = *(const v2f*)&arow1[k + (half << 1)];
            const float* bb = Bp + (size_t)kg * (NPAD * 4) + (half << 1);
            const v2f b0 = *(const v2f*)&bb[col0 * 4];
            const v2f b1 = *(const v2f*)&bb[col1 * 4];
            c00 = __builtin_amdgcn_wmma_f32_16x16x4_f32(false, a0, false, b0, (short)0, c00, false, false);
            c01 = __builtin_amdgcn_wmma_f32_16x16x4_f32(false, a0, false, b1, (short)0, c01, false, false);
            c10 = __builtin_amdgcn_wmma_f32_16x16x4_f32(false, a1, false, b0, (short)0, c10, false, false);
            c11 = __builtin_amdgcn_wmma_f32_16x16x4_f32(false, a1, false, b1, (short)0, c11, false, false);
        }
        // dQd partial: C[r] of M-tile mt holds row (mt*16 + r + half*8),
        // col (n0 + l15). Padded cols read 0 from LDS.
#pragma unroll
        for (int r = 0; r < 8; ++r) {
            const int rr0 = (2 * mp) * 16 + r + (half << 3);
            const int rr1 = rr0 + 16;
            acc0[r] += c00[r] * lds_d[rr0 * LDSTR + col0];
            acc0[r] += c01[r] * lds_d[rr0 * LDSTR + col1];
            acc1[r] += c10[r] * lds_d[rr1 * LDSTR + col0];
            acc1[r] += c11[r] * lds_d[rr1 * LDSTR + col1];
        }
    }

    // Reduce over the 16 lanes of each half-wave (rows stay separated).
#pragma unroll
    for (int r = 0; r < 8; ++r) {
        float v0 = acc0[r], v1 = acc1[r];
#pragma unroll
        for (int o = 1; o < 16; o <<= 1) {
            v0 += __shfl_xor(v0, o, 32);
            v1 += __shfl_xor(v1, o, 32);
        }
        acc0[r] = v0; acc1[r] = v1;
    }
    if (l15 == 0) {
#pragma unroll
        for (int r = 0; r < 8; ++r) {
            lds_part[(wave * 2 + 0) * 16 + (half << 3) + r] = acc0[r];
            lds_part[(wave * 2 + 1) * 16 + (half << 3) + r] = acc1[r];
        }
    }
    __syncthreads();

    // Fixed-order cross-wave combine: thread t -> row0 + (t>>4)*16 + (t&15).
    if (threadIdx.x < 64) {
        const int mt  = threadIdx.x >> 4;     // M-tile 0..3
        const int r   = threadIdx.x & 15;
        const int row = row0 + mt * 16 + r;
        if (row < B) {
            const int mp2 = mt >> 1;          // which wave M-pair
            const int sub = mt & 1;           // which tile inside the pair
            float s = 0.f;
#pragma unroll
            for (int q = 0; q < 4; ++q) {     // the 4 N-column waves
                const int w = mp2 * 4 + q;
                s += lds_part[(w * 2 + sub) * 16 + r];
            }
            ws_dqd[row] = s;
        }
    }
}

// ---------------------------------------------------------------------------
// Kernel 4: epilogue -> fea_probs.
// ---------------------------------------------------------------------------
__global__ __launch_bounds__(256) void finalize_kernel(
    const float* __restrict__ ws_tot, const float* __restrict__ ws_l2,
    const float* __restrict__ ws_dqd, const float* __restrict__ ws_scalar,
    float* __restrict__ out, int n)
{
    const int i = blockIdx.x * 256 + threadIdx.x;
    if (i >= n) return;
    const float lsc = ws_scalar[0];
    const float dQd = ws_dqd[i];
    float tot = ws_tot[i];
    tot += relu_(0.6f - lsc);
    tot += relu_(dQd - 0.005f) * 0.5f + relu_(0.0025f - dQd) * 0.5f;           // W_11
    tot += relu_(100.0f * dQd - 100.0f * ws_l2[i] - 1000.0f) * 10.0f;          // ZSTAR
    out[i] = relu_(1.0f - tanhf(tot * 0.01f));                                 // INF_SCALE
}

extern "C" void kernel_launch(void* const* d_in, const int* in_sizes, int n_in,
                              void* d_out, int out_size, void* d_ws, size_t ws_size,
                              hipStream_t stream) {
    const float* x      = (const float*)d_in[0];
    const float* x_bw   = (const float*)d_in[1];
    const float* alpha  = (const float*)d_in[2];
    const float* beta   = (const float*)d_in[3];
    /* d_in[4] = w_pre : unused by the reference */
    const float* Omega  = (const float*)d_in[5];
    const float* sector = (const float*)d_in[6];
    const float* mq     = (const float*)d_in[7];
    float* out = (float*)d_out;

    const int B = in_sizes[0] / DIM;

    float* ws        = (float*)d_ws;
    float* Bp        = ws;                          // PACKN floats (16B aligned)
    float* ws_tot    = ws + PACKN;
    float* ws_l2     = ws_tot + B;
    float* ws_sa     = ws_l2 + B;
    float* ws_dqd    = ws_sa + B;
    float* ws_scalar = ws_dqd + B;

    pack_omega_kernel<<<(PACKN + 255) / 256, 256, 0, stream>>>(Omega, Bp);
    row_stats_kernel<<<(B + 7) / 8, 256, 0, stream>>>(
        x, x_bw, alpha, beta, sector, mq, ws_tot, ws_l2, ws_sa, B);
    reduce_sa_kernel<<<1, 1024, 0, stream>>>(ws_sa, ws_scalar, B);

    const size_t lds_bytes = (size_t)(64 * LDSTR + 8 * 2 * 16) * sizeof(float);
    gemm_dqd_kernel<<<(B + 63) / 64, 256, lds_bytes, stream>>>(
        x, x_bw, Bp, ws_dqd, B);

    finalize_kernel<<<(B + 255) / 256, 256, 0, stream>>>(
        ws_tot, ws_l2, ws_dqd, ws_scalar, out, B);
}